// Decoder_84069689852145
// MI455X (gfx1250) — compile-verified
//
#include <hip/hip_runtime.h>

typedef __attribute__((ext_vector_type(2))) float v2f;
typedef __attribute__((ext_vector_type(4))) float v4f;
typedef __attribute__((ext_vector_type(8))) float v8f;

#define NN 100000
#define CC 128
#define EE 600000
#define HH 64

#define EDGES_PER_BLOCK 128   // 8 waves x 16 edges
#define W1P_PAD  160          // 128 + 32 dwords -> upper half-wave shifted 32 banks
#define PROD_PAD 132          // 128 + 4 -> per-row A reads hit distinct banks

__global__ __launch_bounds__(256) void decoder_fused(
    const float* __restrict__ z_in,  const float* __restrict__ z_out,
    const float* __restrict__ z_self, const int* __restrict__ ei,
    const float* __restrict__ W1, const float* __restrict__ B1,
    const float* __restrict__ W2, const float* __restrict__ B2,
    float* __restrict__ out)
{
    // W1 K-pair interleaved: sW1p[k>>1][2*n + (k&1)] = W1[k][n]
    // -> B fragment {W1[k][n], W1[k+1][n]} is one aligned ds_load_b64
    __shared__ __align__(16) float sW1p[CC / 2][W1P_PAD]; // 40 KB
    __shared__ __align__(16) float sProd[8][16][PROD_PAD];// 66 KB: A tiles per wave
    __shared__ float sRed[8][16][17];                     // 8.5 KB reduction scratch
    __shared__ float sB1[HH];
    __shared__ float sW2[HH];
    __shared__ int   sSrc[EDGES_PER_BLOCK];
    __shared__ int   sDst[EDGES_PER_BLOCK];

    const int tid  = threadIdx.x;
    const int wave = tid >> 5;
    const int lane = tid & 31;
    const int blockBase = blockIdx.x * EDGES_PER_BLOCK;

    // ---- Stage W1 (128x64 fp32) cooperatively, K-pair interleaved ----
    #pragma unroll
    for (int j = 0; j < 4; ++j) {
        int idx = tid + j * 256;          // 0..1023 = 64 pair-rows x 16 h-chunks
        int pr  = idx >> 4;               // pair-row: covers K = 2pr, 2pr+1
        int h0  = (idx & 15) * 4;
        v4f wa = *(const v4f*)(W1 + (2 * pr)     * HH + h0);
        v4f wb = *(const v4f*)(W1 + (2 * pr + 1) * HH + h0);
        v2f p0 = {wa.x, wb.x}; v2f p1 = {wa.y, wb.y};
        v2f p2 = {wa.z, wb.z}; v2f p3 = {wa.w, wb.w};
        *(v2f*)&sW1p[pr][2 * (h0 + 0)] = p0;
        *(v2f*)&sW1p[pr][2 * (h0 + 1)] = p1;
        *(v2f*)&sW1p[pr][2 * (h0 + 2)] = p2;
        *(v2f*)&sW1p[pr][2 * (h0 + 3)] = p3;
    }
    if (tid < HH) { sB1[tid] = B1[tid]; sW2[tid] = W2[tid]; }
    if (tid < EDGES_PER_BLOCK) {
        int e = blockBase + tid; if (e >= EE) e = EE - 1;   // clamp: no divergence later
        int s = ei[e];      s = s < 0 ? 0 : (s >= NN ? NN - 1 : s);
        int d = ei[EE + e]; d = d < 0 ? 0 : (d >= NN ? NN - 1 : d);
        sSrc[tid] = s; sDst[tid] = d;
    }
    __syncthreads();

    // ---- Gather + fused elementwise: A[m][:] = (z_out[s]+z_self[s])*(z_in[d]+z_self[d]) ----
    // One fully-coalesced 512B row per wave-iteration (32 lanes x float4); L2-resident.
    const int col = lane * 4;
    #pragma unroll 4
    for (int m = 0; m < 16; ++m) {
        int em = wave * 16 + m;
        size_t sOff = (size_t)sSrc[em] * CC + col;
        size_t dOff = (size_t)sDst[em] * CC + col;
        v4f a  = *(const v4f*)(z_out  + sOff);
        v4f as = *(const v4f*)(z_self + sOff);
        v4f b  = *(const v4f*)(z_in   + dOff);
        v4f bs = *(const v4f*)(z_self + dOff);
        v4f p  = (a + as) * (b + bs);
        *(v4f*)&sProd[wave][m][col] = p;
    }
    // same-wave DS ops are in-order: sProd store->load needs no barrier

    // ---- GEMM: D[16x64] = A[16x128] x W1[128x64] via v_wmma_f32_16x16x4_f32 ----
    // A 16x4 f32 layout: lanes 0-15 (M=lane) hold K=k0,k0+1 ; lanes 16-31 hold K=k0+2,k0+3
    const int arow = lane & 15;
    const int half = lane >> 4;                 // 0 or 1
    const float* aptr = &sProd[wave][arow][half << 1];
    const float* bptr = &sW1p[half][2 * arow];  // pair-row stride below is immediate

    v8f zero8 = {0.f,0.f,0.f,0.f,0.f,0.f,0.f,0.f};
    v8f acc[4] = {zero8, zero8, zero8, zero8};  // 4 N-tiles of 16

    #pragma unroll 8
    for (int i = 0; i < 32; ++i) {
        v2f afrag = *(const v2f*)(aptr + 4 * i);                 // ds_load_b64
        #pragma unroll
        for (int t = 0; t < 4; ++t) {
            v2f bfrag = *(const v2f*)(bptr + 2 * i * W1P_PAD + 32 * t); // ds_load_b64, imm offset
            acc[t] = __builtin_amdgcn_wmma_f32_16x16x4_f32(
                false, afrag, false, bfrag, (short)0, acc[t], false, false);
        }
    }

    // ---- Epilogue: +b1, ELU, dot with W2 (partial per lane), LDS reduce, sigmoid ----
    // C/D layout: lane<16 -> N=lane, VGPR r -> M=r ; lane>=16 -> N=lane-16, M=r+8
    const int mbase = half * 8;
    #pragma unroll
    for (int r = 0; r < 8; ++r) {
        float partial = 0.f;
        #pragma unroll
        for (int t = 0; t < 4; ++t) {
            int n = t * 16 + arow;
            float x  = acc[t][r] + sB1[n];
            float hx = x > 0.f ? x : (__expf(x) - 1.f);   // ELU
            partial += hx * sW2[n];
        }
        sRed[wave][mbase + r][arow] = partial;
    }
    __syncthreads();

    if (lane < 16) {
        int e = blockBase + wave * 16 + lane;
        if (e < EE) {
            float s = 0.f;
            #pragma unroll
            for (int c2 = 0; c2 < 16; ++c2) s += sRed[wave][lane][c2];
            float logit = s + B2[0];
            out[e] = 1.0f / (1.0f + __expf(-logit));      // sigmoid
        }
    }
}

extern "C" void kernel_launch(void* const* d_in, const int* in_sizes, int n_in,
                              void* d_out, int out_size, void* d_ws, size_t ws_size,
                              hipStream_t stream) {
    (void)in_sizes; (void)n_in; (void)d_ws; (void)ws_size; (void)out_size;
    const float* z_in   = (const float*)d_in[0];
    const float* z_out  = (const float*)d_in[1];
    const float* z_self = (const float*)d_in[2];
    const int*   ei     = (const int*)d_in[3];
    const float* W1     = (const float*)d_in[4];
    const float* B1     = (const float*)d_in[5];
    const float* W2     = (const float*)d_in[6];
    const float* B2     = (const float*)d_in[7];
    float* out = (float*)d_out;

    dim3 grid((EE + EDGES_PER_BLOCK - 1) / EDGES_PER_BLOCK);  // 4688 blocks
    decoder_fused<<<grid, 256, 0, stream>>>(z_in, z_out, z_self, ei, W1, B1, W2, B2, out);
}